// Stage2GNN_34093450396551
// MI455X (gfx1250) — compile-verified
//
#include <hip/hip_runtime.h>
#include <math.h>

#define N_NODES 40000
#define N_EDGES 640000
#define IN_DIM  128
#define SAT_DIM 64
#define HID     128
#define OUT_DIM 54
#define NEG_SLOPE 0.2f

typedef __attribute__((ext_vector_type(2))) float v2f;
typedef __attribute__((ext_vector_type(8))) float v8f;

// ---------- helpers ----------
__device__ __forceinline__ unsigned f2ord(float f) {
    unsigned b = __float_as_uint(f);
    return (b & 0x80000000u) ? ~b : (b | 0x80000000u);
}
__device__ __forceinline__ float ord2f(unsigned u) {
    unsigned b = (u & 0x80000000u) ? (u & 0x7fffffffu) : ~u;
    return __uint_as_float(b);
}
#define ORD_NEG_INF 0x007FFFFFu   // f2ord(-inf)

// ---------- fp32 WMMA GEMM, full-width: C[M,N] = A[M,K] @ W[N,K]^T (+bias)(+relu) ----------
// N must be a multiple of 32. One wave computes a 16x32 tile (2 accumulators),
// reusing the A fragment across both WMMAs. No per-lane masking in the k-loop.
__global__ void gemm_wmma_full(const float* __restrict__ A, const float* __restrict__ W,
                               const float* __restrict__ bias, float* __restrict__ C,
                               int M, int N, int K, int ntilesN32, int relu) {
    int wid  = (blockIdx.x * blockDim.x + threadIdx.x) >> 5;
    int lane = threadIdx.x & 31;
    int totalTiles = (M >> 4) * ntilesN32;
    if (wid >= totalTiles) return;           // wave-uniform: EXEC full at every WMMA
    int tm = wid / ntilesN32, tn = wid % ntilesN32;
    int half = lane >> 4, l16 = lane & 15;
    int mrow  = tm * 16 + l16;
    int ncol0 = tn * 32 + l16;
    const float* arow  = A + (size_t)mrow  * K;
    const float* wrow0 = W + (size_t)ncol0 * K;
    const float* wrow1 = wrow0 + (size_t)16 * K;

    v8f acc0 = {}, acc1 = {};
    for (int k = 0; k < K; k += 4) {
        int kk = k + 2 * half;
        v2f a  = *(const v2f*)(arow  + kk);
        v2f b0 = *(const v2f*)(wrow0 + kk);
        v2f b1 = *(const v2f*)(wrow1 + kk);
        acc0 = __builtin_amdgcn_wmma_f32_16x16x4_f32(false, a, false, b0,
                                                     (short)0, acc0, false, false);
        acc1 = __builtin_amdgcn_wmma_f32_16x16x4_f32(false, a, false, b1,
                                                     (short)0, acc1, false, false);
    }
    float bv0 = bias ? bias[ncol0]      : 0.0f;
    float bv1 = bias ? bias[ncol0 + 16] : 0.0f;
#pragma unroll
    for (int r = 0; r < 8; ++r) {
        int row = tm * 16 + r + 8 * half;
        float v0 = acc0[r] + bv0;
        float v1 = acc1[r] + bv1;
        if (relu) { v0 = fmaxf(v0, 0.0f); v1 = fmaxf(v1, 0.0f); }
        C[(size_t)row * N + ncol0]      = v0;
        C[(size_t)row * N + ncol0 + 16] = v1;
    }
}

// ---------- fp32 WMMA GEMM, lane-masked (ragged N, e.g. fc2's N=54) ----------
__global__ void gemm_wmma_masked(const float* __restrict__ A, const float* __restrict__ W,
                                 const float* __restrict__ bias, float* __restrict__ C,
                                 int M, int N, int K, int ntilesN, int relu) {
    int wid  = (blockIdx.x * blockDim.x + threadIdx.x) >> 5;
    int lane = threadIdx.x & 31;
    int totalTiles = (M >> 4) * ntilesN;
    if (wid >= totalTiles) return;
    int tm = wid / ntilesN, tn = wid % ntilesN;
    int half = lane >> 4, l16 = lane & 15;
    int mrow = tm * 16 + l16;
    int ncol = tn * 16 + l16;
    bool nvalid = (ncol < N);
    int ncl = nvalid ? ncol : (N - 1);       // clamp so B loads are always in-bounds
    const float* arow = A + (size_t)mrow * K;
    const float* wrow = W + (size_t)ncl * K;

    v8f acc = {};
    for (int k = 0; k < K; k += 4) {
        int kk = k + 2 * half;
        v2f a = *(const v2f*)(arow + kk);
        v2f braw = *(const v2f*)(wrow + kk);
        v2f b;
        b.x = nvalid ? braw.x : 0.0f;
        b.y = nvalid ? braw.y : 0.0f;
        acc = __builtin_amdgcn_wmma_f32_16x16x4_f32(false, a, false, b,
                                                    (short)0, acc, false, false);
    }
#pragma unroll
    for (int r = 0; r < 8; ++r) {
        int row = tm * 16 + r + 8 * half;
        if (nvalid) {
            float v = acc[r];
            if (bias) v += bias[ncol];
            if (relu) v = fmaxf(v, 0.0f);
            C[(size_t)row * N + ncol] = v;
        }
    }
}

// ---------- fused input: h = relu(x[:,:64]@satW^T + satB) + 0.5*relu(x[:,64:]@neiW^T + neiB) ----------
__global__ void input_fuse_wmma(const float* __restrict__ x,
                                const float* __restrict__ satW, const float* __restrict__ satB,
                                const float* __restrict__ neiW, const float* __restrict__ neiB,
                                float* __restrict__ H) {
    int wid  = (blockIdx.x * blockDim.x + threadIdx.x) >> 5;
    int lane = threadIdx.x & 31;
    const int ntilesN = HID / 16;
    int totalTiles = (N_NODES / 16) * ntilesN;
    if (wid >= totalTiles) return;
    int tm = wid / ntilesN, tn = wid % ntilesN;
    int half = lane >> 4, l16 = lane & 15;
    int mrow = tm * 16 + l16;
    int ncol = tn * 16 + l16;
    const float* xrow = x + (size_t)mrow * IN_DIM;
    const float* sw   = satW + (size_t)ncol * SAT_DIM;
    const float* nw   = neiW + (size_t)ncol * SAT_DIM;

    v8f acc1 = {}, acc2 = {};
    for (int k = 0; k < SAT_DIM; k += 4) {
        int kk = k + 2 * half;
        v2f a1 = *(const v2f*)(xrow + kk);
        v2f b1 = *(const v2f*)(sw + kk);
        v2f a2 = *(const v2f*)(xrow + SAT_DIM + kk);
        v2f b2 = *(const v2f*)(nw + kk);
        acc1 = __builtin_amdgcn_wmma_f32_16x16x4_f32(false, a1, false, b1,
                                                     (short)0, acc1, false, false);
        acc2 = __builtin_amdgcn_wmma_f32_16x16x4_f32(false, a2, false, b2,
                                                     (short)0, acc2, false, false);
    }
#pragma unroll
    for (int r = 0; r < 8; ++r) {
        int row = tm * 16 + r + 8 * half;
        float s = fmaxf(acc1[r] + satB[ncol], 0.0f);
        float n = fmaxf(acc2[r] + neiB[ncol], 0.0f);
        H[(size_t)row * HID + ncol] = s + 0.5f * n;
    }
}

// ---------- per-node attention dots: as[n] = h1[n]·a_src, ad[n] = h1[n]·a_dst ----------
__global__ void alpha_kernel(const float* __restrict__ h1,
                             const float* __restrict__ a_src, const float* __restrict__ a_dst,
                             float* __restrict__ as, float* __restrict__ ad) {
    int wid  = (blockIdx.x * blockDim.x + threadIdx.x) >> 5;
    int lane = threadIdx.x & 31;
    if (wid >= N_NODES) return;
    const float* hr = h1 + (size_t)wid * HID;
    float s = 0.0f, d = 0.0f;
#pragma unroll
    for (int c = lane; c < HID; c += 32) {
        float v = hr[c];
        s += v * a_src[c];
        d += v * a_dst[c];
    }
#pragma unroll
    for (int off = 16; off > 0; off >>= 1) {
        s += __shfl_xor(s, off, 32);
        d += __shfl_xor(d, off, 32);
    }
    if (lane == 0) { as[wid] = s; ad[wid] = d; }
}

// ---------- zero accumulators ----------
__global__ void init_kernel(float* __restrict__ acc, float* __restrict__ denom,
                            unsigned* __restrict__ m_u) {
    int i = blockIdx.x * blockDim.x + threadIdx.x;
    if (i < N_NODES * HID) acc[i] = 0.0f;
    if (i < N_NODES) { denom[i] = 0.0f; m_u[i] = ORD_NEG_INF; }
}

// ---------- segment max over incoming edges (monotone-uint atomicMax) ----------
__global__ void edge_max_kernel(const int* __restrict__ ei,
                                const float* __restrict__ as, const float* __restrict__ ad,
                                unsigned* __restrict__ m_u) {
    int i = blockIdx.x * blockDim.x + threadIdx.x;
    int total = N_EDGES + N_NODES;
    if (i >= total) return;
    int s, d;
    if (i < N_EDGES) { s = ei[i]; d = ei[N_EDGES + i]; }
    else             { s = d = i - N_EDGES; }
    float e = as[s] + ad[d];
    e = (e >= 0.0f) ? e : NEG_SLOPE * e;
    atomicMax(&m_u[d], f2ord(e));
}

// ---------- weighted scatter: denom[d]+=w ; acc[d,:]+=w*h1[s,:]  (one wave per edge) ----------
__global__ void edge_acc_kernel(const int* __restrict__ ei,
                                const float* __restrict__ as, const float* __restrict__ ad,
                                const unsigned* __restrict__ m_u,
                                const float* __restrict__ h1,
                                float* __restrict__ denom, float* __restrict__ acc) {
    int wid  = (blockIdx.x * blockDim.x + threadIdx.x) >> 5;
    int lane = threadIdx.x & 31;
    int total = N_EDGES + N_NODES;
    if (wid >= total) return;
    int s, d;
    if (wid < N_EDGES) { s = ei[wid]; d = ei[N_EDGES + wid]; }
    else               { s = d = wid - N_EDGES; }
    float e = as[s] + ad[d];
    e = (e >= 0.0f) ? e : NEG_SLOPE * e;
    float w = expf(e - ord2f(m_u[d]));
    if (lane == 0) atomicAdd(&denom[d], w);
    const float* hs = h1 + (size_t)s * HID;
    float* ac = acc + (size_t)d * HID;
#pragma unroll
    for (int c = lane; c < HID; c += 32)
        atomicAdd(&ac[c], w * hs[c]);
}

// ---------- finalize: h = relu(acc/denom + bias) ----------
__global__ void finalize_kernel(const float* __restrict__ acc, const float* __restrict__ denom,
                                const float* __restrict__ bias, float* __restrict__ out) {
    int i = blockIdx.x * blockDim.x + threadIdx.x;
    if (i >= N_NODES * HID) return;
    int n = i >> 7, c = i & (HID - 1);
    float v = acc[i] / (denom[n] + 1e-16f) + bias[c];
    out[i] = fmaxf(v, 0.0f);
}

extern "C" void kernel_launch(void* const* d_in, const int* in_sizes, int n_in,
                              void* d_out, int out_size, void* d_ws, size_t ws_size,
                              hipStream_t stream) {
    (void)in_sizes; (void)n_in; (void)out_size; (void)ws_size;
    const float* x          = (const float*)d_in[0];
    const int*   edge_index = (const int*)  d_in[1];
    const float* sat_W  = (const float*)d_in[2];
    const float* sat_b  = (const float*)d_in[3];
    const float* nei_W  = (const float*)d_in[4];
    const float* nei_b  = (const float*)d_in[5];
    const float* conv_W[2]    = {(const float*)d_in[6],  (const float*)d_in[10]};
    const float* conv_asrc[2] = {(const float*)d_in[7],  (const float*)d_in[11]};
    const float* conv_adst[2] = {(const float*)d_in[8],  (const float*)d_in[12]};
    const float* conv_b[2]    = {(const float*)d_in[9],  (const float*)d_in[13]};
    const float* fc1_W = (const float*)d_in[14];
    const float* fc1_b = (const float*)d_in[15];
    const float* fc2_W = (const float*)d_in[16];
    const float* fc2_b = (const float*)d_in[17];
    float* out = (float*)d_out;

    const size_t NH = (size_t)N_NODES * HID;
    float* bufA  = (float*)d_ws;          // node features h
    float* bufB  = bufA + NH;             // h' = h @ W^T
    float* bufC  = bufB + NH;             // scatter accumulator
    float* as    = bufC + NH;
    float* ad    = as + N_NODES;
    float* denom = ad + N_NODES;
    unsigned* m_u = (unsigned*)(denom + N_NODES);

    const int THR = 128;                  // 4 waves/block for GEMM kernels
    // full GEMM (N=128): 16x32 tiles -> (40000/16)*(128/32)=10000 tiles -> 2500 blocks
    const int fullTiles  = (N_NODES / 16) * (HID / 32);
    const int fullBlocks = (fullTiles + 3) / 4;
    // input fuse: 16x16 tiles -> 20000 tiles -> 5000 blocks
    const int fuseBlocks = ((N_NODES / 16) * (HID / 16) + 3) / 4;
    const int elemBlocks = (N_NODES * HID + 255) / 256;
    const int ET = N_EDGES + N_NODES;                          // 680000
    const int edgeThrBlocks  = (ET + 255) / 256;
    const int edgeWaveBlocks = (ET + 7) / 8;                   // 8 waves/256-thr block

    // Stage 0: fused input projection
    input_fuse_wmma<<<fuseBlocks, THR, 0, stream>>>(x, sat_W, sat_b, nei_W, nei_b, bufA);

    // Stages 1-2: GAT layers
    for (int L = 0; L < 2; ++L) {
        gemm_wmma_full<<<fullBlocks, THR, 0, stream>>>(bufA, conv_W[L], nullptr, bufB,
                                                       N_NODES, HID, HID, HID / 32, 0);
        alpha_kernel<<<(N_NODES + 7) / 8, 256, 0, stream>>>(bufB, conv_asrc[L], conv_adst[L], as, ad);
        init_kernel<<<elemBlocks, 256, 0, stream>>>(bufC, denom, m_u);
        edge_max_kernel<<<edgeThrBlocks, 256, 0, stream>>>(edge_index, as, ad, m_u);
        edge_acc_kernel<<<edgeWaveBlocks, 256, 0, stream>>>(edge_index, as, ad, m_u, bufB, denom, bufC);
        finalize_kernel<<<elemBlocks, 256, 0, stream>>>(bufC, denom, conv_b[L], bufA);
    }

    // fc1 + relu
    gemm_wmma_full<<<fullBlocks, THR, 0, stream>>>(bufA, fc1_W, fc1_b, bufB,
                                                   N_NODES, HID, HID, HID / 32, 1);
    // fc2 (N=54, padded to 4 N-tiles with lane masking)
    const int fc2Tiles = (N_NODES / 16) * 4;                   // 10000
    gemm_wmma_masked<<<(fc2Tiles + 3) / 4, THR, 0, stream>>>(bufB, fc2_W, fc2_b, out,
                                                             N_NODES, OUT_DIM, HID, 4, 0);
}